// KNN_79345225826553
// MI455X (gfx1250) — compile-verified
//
#include <hip/hip_runtime.h>
#include <hip/hip_bf16.h>
#include <stdint.h>

typedef __attribute__((ext_vector_type(2))) float v2f;
typedef __attribute__((ext_vector_type(8))) float v8f;

#define NN    1024   // nodes
#define NU    256    // unknown nodes
#define ROWS  64     // (b,t) rows per workgroup
#define KC    64     // K chunk
#define KCP   65     // padded row stride for sX/sW (conflict-free LDS)
#define NUP   257    // padded row stride for sY
#define NT    256    // threads per block (8 wave32)

// smem main region: GEMM staging (sX: ROWS*KCP, sW: NU*KCP) reused as sY (ROWS*NUP)
#define SMEM_MAIN 20800  // max(64*65 + 256*65, 64*257)

__global__ __launch_bounds__(NT, 1)
void graph_impute_kernel(const float* __restrict__ x,
                         const float* __restrict__ A,
                         const int*   __restrict__ unk,
                         const float* __restrict__ maskp,
                         float*       __restrict__ out,
                         int BT)
{
    __shared__ float    smem[SMEM_MAIN];
    __shared__ float    sRdeg[NU];
    __shared__ float    sBase[NU];
    __shared__ unsigned sLbits[NU][NU / 32];
    __shared__ int      sUnk[NU];
    __shared__ short    sIdx[NN];     // node -> unknown index, -1 if known

    const int tid  = threadIdx.x;
    const int lane = tid & 31;
    const int wave = tid >> 5;
    const int rowBase = blockIdx.x * ROWS;
    if (rowBase >= BT) return;
    const float maskv = maskp[0];

    // ---- phase 1: unknown list + node->index map --------------------------
    for (int n = tid; n < NN; n += NT) sIdx[n] = (short)-1;
    if (tid < NU) sUnk[tid] = unk[tid];
    __syncthreads();
    if (tid < NU) sIdx[sUnk[tid]] = (short)tid;
    __syncthreads();

    // ---- phase 2: per-unknown-node stats (wave-cooperative) ---------------
    // deg_i, base_i = mask * sum_{j>=i} A[u_i,u_j], Lbits[i] = bits of A[u_i,u_j]!=0, j<i
    for (int i = wave; i < NU; i += (NT / 32)) {
        const float* arow = A + (size_t)sUnk[i] * NN;
        float deg = 0.f;
        for (int n = lane; n < NN; n += 32) deg += arow[n];
        for (int off = 16; off > 0; off >>= 1) deg += __shfl_xor(deg, off, 32);
        float m = 0.f;
        for (int q = 0; q < NU / 32; ++q) {
            int j = q * 32 + lane;
            float a = arow[sUnk[j]];
            unsigned b = (unsigned)__ballot((a != 0.f) && (j < i));
            if (lane == 0) sLbits[i][q] = b;
            if (j >= i) m += a;
        }
        for (int off = 16; off > 0; off >>= 1) m += __shfl_xor(m, off, 32);
        if (lane == 0) { sRdeg[i] = 1.f / deg; sBase[i] = maskv * m; }
    }

    // ---- phase 3: GEMM  C[64,256] = X[64,1024] x W[1024,256] --------------
    float* sX = smem;              // [ROWS][KCP]
    float* sW = smem + ROWS * KCP; // [NU][KCP]

    const v8f vz = {0.f, 0.f, 0.f, 0.f, 0.f, 0.f, 0.f, 0.f};
    v8f acc[8];
#pragma unroll
    for (int c = 0; c < 8; ++c) acc[c] = vz;

    const int rt   = wave & 3;          // row tile (4 tiles of 16 rows)
    const int ch   = wave >> 2;         // column half (2 halves of 8 col tiles)
    const int mrow = lane & 15;
    const int koff = (lane >> 4) * 2;   // K split across lane halves per ISA layout

    for (int k0 = 0; k0 < NN; k0 += KC) {
        __syncthreads();
        // stage X chunk: 64 rows x 64 K
        for (int idx = tid; idx < ROWS * (KC / 4); idx += NT) {
            int r = idx >> 4, kv = idx & 15;
            float4 v = ((const float4*)(x + (size_t)(rowBase + r) * NN + k0))[kv];
            float* d = &sX[r * KCP + kv * 4];
            d[0] = v.x; d[1] = v.y; d[2] = v.z; d[3] = v.w;
        }
        // stage masked W chunk: W[k,i] = isKnown(k) ? A[u_i,k] : 0   (stored [i][k])
        for (int idx = tid; idx < NU * (KC / 4); idx += NT) {
            int i = idx >> 4, kv = idx & 15;
            float4 a = ((const float4*)(A + (size_t)sUnk[i] * NN + k0))[kv];
            int kb = k0 + kv * 4;
            float* d = &sW[i * KCP + kv * 4];
            d[0] = (sIdx[kb + 0] < 0) ? a.x : 0.f;
            d[1] = (sIdx[kb + 1] < 0) ? a.y : 0.f;
            d[2] = (sIdx[kb + 2] < 0) ? a.z : 0.f;
            d[3] = (sIdx[kb + 3] < 0) ? a.w : 0.f;
        }
        __syncthreads();
        for (int ks = 0; ks < KC; ks += 4) {
            const float* xr = &sX[(rt * 16 + mrow) * KCP + ks + koff];
            v2f af; af.x = xr[0]; af.y = xr[1];
#pragma unroll
            for (int c = 0; c < 8; ++c) {
                const float* wr = &sW[((ch * 8 + c) * 16 + mrow) * KCP + ks + koff];
                v2f bf; bf.x = wr[0]; bf.y = wr[1];
                acc[c] = __builtin_amdgcn_wmma_f32_16x16x4_f32(
                    false, af, false, bf, (short)0, acc[c], false, false);
            }
        }
    }
    __syncthreads();   // done reading sX/sW before aliasing as sY

    // ---- phase 4: accumulators -> LDS (C/D layout: VGPR v = row v / v+8) --
    float* sY = smem;  // [ROWS][NUP]
    {
        const int rb = rt * 16 + (lane >> 4) * 8;
#pragma unroll
        for (int c = 0; c < 8; ++c) {
            int col = (ch * 8 + c) * 16 + mrow;
#pragma unroll
            for (int v = 0; v < 8; ++v)
                sY[(rb + v) * NUP + col] = acc[c][v];
        }
    }
    __syncthreads();

    // ---- phase 5: row-private forward substitution ------------------------
    if (tid < ROWS) {
        float* y = sY + tid * NUP;
        for (int i = 0; i < NU; ++i) {
            float s = y[i] + sBase[i];
            int wmax = i >> 5;
            for (int w = 0; w <= wmax; ++w) {
                unsigned bm = sLbits[i][w];
                while (bm) {
                    int b = __builtin_ctz(bm);
                    bm &= bm - 1;
                    s += y[w * 32 + b];
                }
            }
            y[i] = s * sRdeg[i];
        }
    }
    __syncthreads();

    // ---- phase 6: stream output rows, substituting solved unknowns --------
    for (int idx = tid; idx < ROWS * (NN / 4); idx += NT) {
        int r = idx >> 8, nv = idx & 255;
        size_t gro = (size_t)(rowBase + r) * NN;
        float4 v = ((const float4*)(x + gro))[nv];
        int n = nv * 4;
        short i0 = sIdx[n], i1 = sIdx[n + 1], i2 = sIdx[n + 2], i3 = sIdx[n + 3];
        const float* y = sY + r * NUP;
        if (i0 >= 0) v.x = y[i0];
        if (i1 >= 0) v.y = y[i1];
        if (i2 >= 0) v.z = y[i2];
        if (i3 >= 0) v.w = y[i3];
        ((float4*)(out + gro))[nv] = v;
    }
}

extern "C" void kernel_launch(void* const* d_in, const int* in_sizes, int n_in,
                              void* d_out, int out_size, void* d_ws, size_t ws_size,
                              hipStream_t stream) {
    const float* x       = (const float*)d_in[0];
    const float* A       = (const float*)d_in[1];
    const int*   unknown = (const int*)d_in[2];
    const float* mask    = (const float*)d_in[3];
    float* out = (float*)d_out;

    int BT = in_sizes[0] / NN;               // 8 * 4096 = 32768
    int grid = (BT + ROWS - 1) / ROWS;       // 512 workgroups
    graph_impute_kernel<<<dim3(grid), dim3(NT), 0, stream>>>(x, A, unknown, mask, out, BT);
}